// Model_22720376996402
// MI455X (gfx1250) — compile-verified
//
#include <hip/hip_runtime.h>
#include <hip/hip_bf16.h>

typedef __attribute__((ext_vector_type(2))) float v2f;
typedef __attribute__((ext_vector_type(4))) float v4f;
typedef __attribute__((ext_vector_type(8))) float v8f;

#define VOCAB 128
#define EMB   10
#define HID   20
#define TT    2048

// ---------------------------------------------------------------------------
// Kernel 1: xw[bt][j] = sum_e emb[tok[bt]][e] * W_ih[j][e] + b_ih[j] + b_hh[j]
// Pure bandwidth (writes 42 MB). Weights are wave-uniform -> scalar-cached.
// ---------------------------------------------------------------------------
__global__ void k_embed_proj(const int* __restrict__ inputs,
                             const float* __restrict__ emb,
                             const float* __restrict__ W_ih,
                             const float* __restrict__ b_ih,
                             const float* __restrict__ b_hh,
                             float* __restrict__ xw, int BT) {
    int idx = blockIdx.x * blockDim.x + threadIdx.x;
    if (idx >= BT) return;
    int tok = inputs[idx];
    float x[EMB];
#pragma unroll
    for (int e = 0; e < EMB; ++e) x[e] = emb[tok * EMB + e];
    float o[HID];
#pragma unroll
    for (int j = 0; j < HID; ++j) {
        float acc = b_ih[j] + b_hh[j];
#pragma unroll
        for (int e = 0; e < EMB; ++e) acc = fmaf(x[e], W_ih[j * EMB + e], acc);
        o[j] = acc;
    }
    // 20 floats per row, rows are 80B -> 16B aligned: 5x b128 stores
    float* dst = xw + (size_t)idx * HID;
#pragma unroll
    for (int c = 0; c < HID / 4; ++c) {
        v4f v = {o[4 * c], o[4 * c + 1], o[4 * c + 2], o[4 * c + 3]};
        *(v4f*)(dst + 4 * c) = v;
    }
}

// ---------------------------------------------------------------------------
// Kernel 2: sequential scan. One wave32 per sequence; lane j owns h[j].
// Per-step critical path minimized:
//   tanh (v_tanh_f32, CDNA5 TRANS op) -> v_readlane broadcast (h is
//   wave-uniform per lane index -> SGPR FMA operand) -> 4 parallel
//   accumulator chains of 5 FMAs -> 2 combining adds.
// Next timestep's xw load is issued before the dependent chain (clamped
// index, no branch). The hs store is off the critical path.
// ---------------------------------------------------------------------------
__device__ __forceinline__ float bcast_lane(float v, int lane) {
    return __int_as_float(__builtin_amdgcn_readlane(__float_as_int(v), lane));
}

__global__ void k_scan(const float* __restrict__ xw,
                       const float* __restrict__ W_hh,
                       float* __restrict__ hs) {
    const int b = blockIdx.x;
    const int lane = threadIdx.x;              // 0..31, only 0..19 meaningful
    const int jj = lane < HID ? lane : HID - 1;

    float w[HID];
#pragma unroll
    for (int k = 0; k < HID; ++k) w[k] = W_hh[jj * HID + k];

    const float* xr = xw + (size_t)b * TT * HID;
    float* hr = hs + (size_t)b * TT * HID + lane;

    float h = 0.0f;
    float xc = xr[jj];                         // t = 0
    for (int t = 0; t < TT; ++t) {
        const int tn = (t + 1 < TT) ? (t + 1) : (TT - 1);
        float xn = xr[(size_t)tn * HID + jj];  // prefetch next step (no branch)

        float acc0 = xc, acc1 = 0.0f, acc2 = 0.0f, acc3 = 0.0f;
#pragma unroll
        for (int k = 0; k < 5; ++k) {
            acc0 = fmaf(bcast_lane(h, 4 * k + 0), w[4 * k + 0], acc0);
            acc1 = fmaf(bcast_lane(h, 4 * k + 1), w[4 * k + 1], acc1);
            acc2 = fmaf(bcast_lane(h, 4 * k + 2), w[4 * k + 2], acc2);
            acc3 = fmaf(bcast_lane(h, 4 * k + 3), w[4 * k + 3], acc3);
        }
        float acc = (acc0 + acc1) + (acc2 + acc3);

        float hn;
        asm("v_tanh_f32 %0, %1" : "=v"(hn) : "v"(acc));
        h = hn;
        if (lane < HID) hr[(size_t)t * HID] = h;
        xc = xn;
    }
}

// ---------------------------------------------------------------------------
// Kernel 3: out[BT x 128] = hs[BT x 20] @ W_out^T[20 x 128] + b_out
// V_WMMA_F32_16X16X4_F32, K = 20 = 5 chunks of 4. One wave per 16-row M tile,
// loop over 8 N tiles. Acc preloaded with bias (C layout: lane<16 -> N=lane,
// VGPR r -> M=r; lane>=16 -> N=lane-16, VGPR r -> M=8+r).
// A layout (16x4 f32): lanes 0-15 K={0,1}, lanes 16-31 K={2,3}; B mirrors it.
// ---------------------------------------------------------------------------
__global__ void k_outproj_wmma(const float* __restrict__ hs,
                               const float* __restrict__ W_out,
                               const float* __restrict__ b_out,
                               float* __restrict__ out, int BT) {
    const int lane = threadIdx.x & 31;
    const int wv = threadIdx.x >> 5;
    const int tile = blockIdx.x * 8 + wv;      // 16-row tile index
    const int m_base = tile * 16;
    if (m_base >= BT) return;                  // wave-uniform: EXEC stays all-1

    const int l15 = lane & 15;
    const int klo = (lane >> 4) << 1;          // 0 for lanes 0-15, 2 otherwise
    const int row = m_base + l15;

    // A fragments: 5 K-chunks, each lane holds (K=klo, K=klo+1) of its row.
    // Address = (row*20 + c*4 + klo)*4 bytes -> 8B aligned -> global_load_b64
    v2f a[5];
#pragma unroll
    for (int c = 0; c < 5; ++c)
        a[c] = *(const v2f*)(hs + (size_t)row * HID + c * 4 + klo);

#pragma unroll
    for (int nt = 0; nt < 8; ++nt) {
        const int ncol = nt * 16 + l15;        // output column this lane owns
        v2f bf[5];
#pragma unroll
        for (int c = 0; c < 5; ++c)
            bf[c] = *(const v2f*)(W_out + (size_t)ncol * HID + c * 4 + klo);

        const float bias = b_out[ncol];
        v8f acc = {bias, bias, bias, bias, bias, bias, bias, bias};
#pragma unroll
        for (int c = 0; c < 5; ++c)
            acc = __builtin_amdgcn_wmma_f32_16x16x4_f32(
                false, a[c], false, bf[c], (short)0, acc, false, false);

        const int mrow0 = m_base + ((lane >> 4) << 3);  // +0 or +8
#pragma unroll
        for (int r = 0; r < 8; ++r)
            out[(size_t)(mrow0 + r) * VOCAB + ncol] = acc[r];
    }
}

// ---------------------------------------------------------------------------
extern "C" void kernel_launch(void* const* d_in, const int* in_sizes, int n_in,
                              void* d_out, int out_size, void* d_ws, size_t ws_size,
                              hipStream_t stream) {
    const int*   inputs = (const int*)d_in[0];
    const float* emb    = (const float*)d_in[1];
    const float* W_ih   = (const float*)d_in[2];
    const float* W_hh   = (const float*)d_in[3];
    const float* b_ih   = (const float*)d_in[4];
    const float* b_hh   = (const float*)d_in[5];
    const float* W_out  = (const float*)d_in[6];
    const float* b_out  = (const float*)d_in[7];
    float* out = (float*)d_out;

    const int BT = in_sizes[0];                // B * T = 524288
    const int B  = BT / TT;                    // 256

    float* xw = (float*)d_ws;                  // [BT][20]  (42 MB)
    float* hs = xw + (size_t)BT * HID;         // [BT][20]  (42 MB)

    k_embed_proj<<<(BT + 255) / 256, 256, 0, stream>>>(inputs, emb, W_ih,
                                                       b_ih, b_hh, xw, BT);
    k_scan<<<B, 32, 0, stream>>>(xw, W_hh, hs);

    const int mtiles = BT / 16;                // 32768, 8 waves/block
    k_outproj_wmma<<<mtiles / 8, 256, 0, stream>>>(hs, W_out, b_out, out, BT);
}